// SpectralGCNConv_33818572488735
// MI455X (gfx1250) — compile-verified
//
#include <hip/hip_runtime.h>

typedef __attribute__((ext_vector_type(2))) float v2f;
typedef __attribute__((ext_vector_type(8))) float v8f;

#define D 128          // feature dim (in == out == 128)
#define TILE 16

// ---------------------------------------------------------------------------
// 1) Power iteration for largest singular value, write W_sn = W / sigma.
//    One workgroup of 128 threads; W stays in global (single block, L2-hot).
// ---------------------------------------------------------------------------
__global__ void __launch_bounds__(128)
k_sigma_wsn(const float* __restrict__ W, float* __restrict__ Wsn) {
  __shared__ float v[D], u[D], red[D];
  const int t = threadIdx.x;            // 0..127
  v[t] = 1.0f;
  __syncthreads();
  for (int it = 0; it < 64; ++it) {
    // u = W v   (thread t owns row t)
    float acc = 0.0f;
    for (int k = 0; k < D; ++k) acc += W[t * D + k] * v[k];
    u[t] = acc;
    __syncthreads();
    // w = W^T u (thread t owns column t)
    float acc2 = 0.0f;
    for (int o = 0; o < D; ++o) acc2 += W[o * D + t] * u[o];
    red[t] = acc2 * acc2;
    __syncthreads();
    for (int s = 64; s > 0; s >>= 1) {
      if (t < s) red[t] += red[t + s];
      __syncthreads();
    }
    const float inv = rsqrtf(red[0]);   // 1/||W^T u||
    v[t] = acc2 * inv;
    __syncthreads();
  }
  // sigma = ||W v|| with ||v|| == 1
  float acc = 0.0f;
  for (int k = 0; k < D; ++k) acc += W[t * D + k] * v[k];
  red[t] = acc * acc;
  __syncthreads();
  for (int s = 64; s > 0; s >>= 1) {
    if (t < s) red[t] += red[t + s];
    __syncthreads();
  }
  const float inv_sigma = rsqrtf(red[0]);
  for (int o = 0; o < D; ++o) Wsn[o * D + t] = W[o * D + t] * inv_sigma;
}

// ---------------------------------------------------------------------------
// 2) Degree computation
// ---------------------------------------------------------------------------
__global__ void k_zero_i32(int* __restrict__ p, int n) {
  int i = blockIdx.x * blockDim.x + threadIdx.x;
  if (i < n) p[i] = 0;
}

__global__ void k_deg_count(const int* __restrict__ ei, int* __restrict__ deg, int nE) {
  int i = blockIdx.x * blockDim.x + threadIdx.x;
  if (i < nE) atomicAdd(&deg[ei[nE + i]], 1);   // dst = ei[1][i]
}

__global__ void k_dinv(const int* __restrict__ deg, float* __restrict__ dinv, int n) {
  int i = blockIdx.x * blockDim.x + threadIdx.x;
  if (i < n) dinv[i] = rsqrtf((float)(deg[i] + 1));   // +1 self-loop => deg >= 1
}

// ---------------------------------------------------------------------------
// 3) H = X * Wsn^T via V_WMMA_F32_16X16X4_F32.
//    Block = 256 threads = 8 waves; each wave owns one 16-col output tile of a
//    16-row strip. X strip staged in LDS (row stride 132 to avoid bank
//    conflicts). K loop: 32 x k=4 WMMA steps.
//    ISA layouts (7.12.2):
//      A 16x4 f32 : lanes 0-15 -> M=lane (K=0,1 in VGPR0,1); lanes 16-31 -> K=2,3
//      B 4x16 f32 : lane holds N=lane&15; VGPR0,1 hold K = 2*(lane>>4) + {0,1}
//      C 16x16    : VGPR r -> M = r + 8*(lane>>4), N = lane&15
// ---------------------------------------------------------------------------
__global__ void __launch_bounds__(256)
k_gemm_xwt(const float* __restrict__ X, const float* __restrict__ Wsn,
           float* __restrict__ H, int N) {
  __shared__ float Xs[TILE * 132];
  const int t     = threadIdx.x;
  const int nBase = blockIdx.x * TILE;

  // stage 16x128 strip of X (coalesced global, padded LDS rows)
  for (int i = 0; i < 8; ++i) {
    int lin = t + i * 256;                  // 0..2047
    int r = lin >> 7, c = lin & 127;
    float val = 0.0f;
    if (nBase + r < N) val = X[(size_t)(nBase + r) * D + c];
    Xs[r * 132 + c] = val;
  }
  __syncthreads();

  const int lane  = t & 31;
  const int wv    = t >> 5;                 // 0..7 -> output col tile
  const int oBase = wv * TILE;
  const int mn    = lane & 15;              // M for A, N for B/C
  const int khalf = (lane >> 4) << 1;       // 0 or 2
  const float* __restrict__ Brow = Wsn + (size_t)(oBase + mn) * D;

  v8f c = {0.f, 0.f, 0.f, 0.f, 0.f, 0.f, 0.f, 0.f};
  for (int kb = 0; kb < D; kb += 4) {
    v2f a, b;
    a.x = Xs[mn * 132 + kb + khalf];
    a.y = Xs[mn * 132 + kb + khalf + 1];
    b.x = Brow[kb + khalf];
    b.y = Brow[kb + khalf + 1];
    c = __builtin_amdgcn_wmma_f32_16x16x4_f32(false, a, false, b,
                                              (short)0, c, false, false);
  }

  const int rowOff = (lane >> 4) * 8;
  for (int r = 0; r < 8; ++r) {
    int row = nBase + rowOff + r;
    if (row < N) H[(size_t)row * D + oBase + mn] = c[r];
  }
}

// ---------------------------------------------------------------------------
// 4) out = b + H * dinv^2   (self-loop message, no atomics)
// ---------------------------------------------------------------------------
__global__ void __launch_bounds__(256)
k_outinit(const float* __restrict__ H, const float* __restrict__ dinv,
          const float* __restrict__ bias, float* __restrict__ out, int N) {
  size_t i = (size_t)blockIdx.x * 256 + threadIdx.x;   // over N*128
  if (i >= (size_t)N * D) return;
  int n = (int)(i >> 7);
  int o = (int)(i & 127);
  float di = dinv[n];
  out[i] = bias[o] + H[i] * di * di;
}

// ---------------------------------------------------------------------------
// 5) Edge scatter: one wave per edge; lane loads float4 of h[src] (512B
//    coalesced row) and issues 4 hardware fp32 global atomics to out[dst].
// ---------------------------------------------------------------------------
__global__ void __launch_bounds__(256)
k_scatter(const int* __restrict__ ei, const float* __restrict__ H,
          const float* __restrict__ dinv, float* __restrict__ out, int nE) {
  const int lane = threadIdx.x & 31;
  const int e = blockIdx.x * 8 + (threadIdx.x >> 5);
  if (e >= nE) return;
  const int s = ei[e];
  const int d = ei[nE + e];
  const float coeff = dinv[s] * dinv[d];
  const float4 hv = ((const float4*)(H + (size_t)s * D))[lane];
  float* orow = out + (size_t)d * D + lane * 4;
  unsafeAtomicAdd(orow + 0, hv.x * coeff);
  unsafeAtomicAdd(orow + 1, hv.y * coeff);
  unsafeAtomicAdd(orow + 2, hv.z * coeff);
  unsafeAtomicAdd(orow + 3, hv.w * coeff);
}

// ---------------------------------------------------------------------------
extern "C" void kernel_launch(void* const* d_in, const int* in_sizes, int n_in,
                              void* d_out, int out_size, void* d_ws, size_t ws_size,
                              hipStream_t stream) {
  const float* X  = (const float*)d_in[0];   // [N,128]
  const int*   EI = (const int*)d_in[1];     // [2,E] row-major: src then dst
  const float* W  = (const float*)d_in[2];   // [128,128]
  const float* B  = (const float*)d_in[3];   // [128]
  float* out = (float*)d_out;

  const int N  = in_sizes[0] / D;
  const int nE = in_sizes[1] / 2;

  // workspace layout
  char* ws = (char*)d_ws;
  float* Wsn  = (float*)(ws);                      // 64 KB
  int*   deg  = (int*)  (ws + (64 << 10));         // N * 4 B
  float* dinv = (float*)(ws + (576 << 10));        // N * 4 B
  float* H    = (float*)(ws + (2u << 20));         // N * 128 * 4 B

  k_sigma_wsn<<<1, 128, 0, stream>>>(W, Wsn);
  k_zero_i32<<<(N + 255) / 256, 256, 0, stream>>>(deg, N);
  k_deg_count<<<(nE + 255) / 256, 256, 0, stream>>>(EI, deg, nE);
  k_dinv<<<(N + 255) / 256, 256, 0, stream>>>(deg, dinv, N);
  k_gemm_xwt<<<(N + TILE - 1) / TILE, 256, 0, stream>>>(X, Wsn, H, N);
  {
    size_t tot = (size_t)N * D;
    k_outinit<<<(unsigned)((tot + 255) / 256), 256, 0, stream>>>(H, dinv, B, out, N);
  }
  k_scatter<<<(nE + 7) / 8, 256, 0, stream>>>(EI, H, dinv, out, nE);
}